// PostProcess_29884382445817
// MI455X (gfx1250) — compile-verified
//
#include <hip/hip_runtime.h>
#include <stdint.h>

// Problem constants (from reference setup_inputs)
#define BATCH 64
#define QDET  1000
#define CLS   1203
#define N_ELEM (QDET*CLS)             // 1203000 floats per batch
#define NV4   (N_ELEM/4)              // 300750 float4 per batch (exact)
#define TOPK  100
#define BINS  4096                    // top-12-bit radix histogram
#define CAP   2048                    // candidate cap per batch
#define BPB   64                      // streaming blocks per batch
#define TPB   256                     // threads per block (8 waves, wave32)
#define NTILES ((NV4 + TPB - 1)/TPB)  // 1175 tiles of 256 float4 (1024 floats)
#define MAXITER 32                    // >= ceil(NTILES/BPB) = 19

// ---- order-preserving float <-> uint key transform (sigmoid is monotonic,
// so top-k on keys == top-k on sigmoid(logits)) ----
__device__ __forceinline__ uint32_t f2key(float f){
  uint32_t u = __float_as_uint(f);
  return u ^ ((uint32_t)((int32_t)u >> 31) | 0x80000000u);
}
__device__ __forceinline__ float key2f(uint32_t k){
  uint32_t bits = (k & 0x80000000u) ? (k ^ 0x80000000u) : ~k;
  return __uint_as_float(bits);
}
__device__ __forceinline__ uint32_t umax4(uint32_t a, uint32_t b, uint32_t c, uint32_t d){
  uint32_t ab = a > b ? a : b;
  uint32_t cd = c > d ? c : d;
  return ab > cd ? ab : cd;
}

// ---- CDNA5 async global->LDS copy (16B per lane), tracked by ASYNCcnt ----
__device__ __forceinline__ void async_copy_b128(uint32_t lds_byte_off, uint64_t gaddr){
  asm volatile("global_load_async_to_lds_b128 %0, %1, off"
               :: "v"(lds_byte_off), "v"(gaddr)
               : "memory");
}
#define WAIT_ASYNC_1() asm volatile("s_wait_asynccnt 0x1" ::: "memory")
#define WAIT_ASYNC_0() asm volatile("s_wait_asynccnt 0x0" ::: "memory")

// ---------------------------------------------------------------------------
// Pass 0: zero histograms + counters (ws is poisoned once, never re-poisoned)
// ---------------------------------------------------------------------------
__global__ void zero_kernel(uint32_t* __restrict__ p, int n){
  int i = blockIdx.x*blockDim.x + threadIdx.x;
  if (i < n) p[i] = 0u;
}

// ---------------------------------------------------------------------------
// Pass 1: per-batch 4096-bin histogram of key top-12-bits, plus per-tile max
// key (enables sparse second pass). Streaming uses per-wave double-buffered
// async global->LDS tiles; the hot loop has no block-wide barriers.
// ---------------------------------------------------------------------------
__global__ void __launch_bounds__(TPB) hist_kernel(const float* __restrict__ logits,
                                                   uint32_t* __restrict__ ghist,
                                                   uint32_t* __restrict__ gtmax){
  __shared__ uint32_t lhist[BINS];
  __shared__ uint32_t tmax[MAXITER];      // per tile-iteration max key
  __shared__ float4 stage[(TPB/32)*64];   // per wave: 2 buffers x 32 lanes x 16B
  const int b   = blockIdx.y;
  const int tid = threadIdx.x;
  for (int i = tid; i < BINS; i += TPB) lhist[i] = 0u;
  if (tid < MAXITER) tmax[tid] = 0u;
  __syncthreads();

  const int wave = tid >> 5;
  const int lane = tid & 31;
  const float4* base = reinterpret_cast<const float4*>(logits) + (size_t)b*NV4;
  float4* wstage = &stage[wave*64];

  int t = blockIdx.x;
  int it = 0;
  if (t < NTILES){
    { // prologue: issue tile t into buf 0 (clamp tail lanes; guarded at use;
      // clamp target NV4-1 lies inside the tail tile, so no cross-tile max leak)
      uint32_t v4 = (uint32_t)t*TPB + (uint32_t)(wave*32 + lane);
      if (v4 >= (uint32_t)NV4) v4 = NV4 - 1;
      async_copy_b128((uint32_t)(uintptr_t)&wstage[lane],
                      (uint64_t)(uintptr_t)(base + v4));
    }
    int buf = 0;
    for (;;){
      int tn = t + BPB;
      if (tn < NTILES){
        uint32_t v4n = (uint32_t)tn*TPB + (uint32_t)(wave*32 + lane);
        if (v4n >= (uint32_t)NV4) v4n = NV4 - 1;
        async_copy_b128((uint32_t)(uintptr_t)&wstage[(buf^1)*32 + lane],
                        (uint64_t)(uintptr_t)(base + v4n));
        WAIT_ASYNC_1();               // tile t landed; tile tn in flight
      } else {
        WAIT_ASYNC_0();               // drain
      }
      uint32_t v4 = (uint32_t)t*TPB + (uint32_t)(wave*32 + lane);
      if (v4 < (uint32_t)NV4){
        float4 v = wstage[buf*32 + lane];
        uint32_t k0 = f2key(v.x), k1 = f2key(v.y), k2 = f2key(v.z), k3 = f2key(v.w);
        atomicAdd(&lhist[k0 >> 20], 1u);
        atomicAdd(&lhist[k1 >> 20], 1u);
        atomicAdd(&lhist[k2 >> 20], 1u);
        atomicAdd(&lhist[k3 >> 20], 1u);
        atomicMax(&tmax[it], umax4(k0, k1, k2, k3));
      }
      if (tn >= NTILES) break;
      t = tn;
      buf ^= 1;
      ++it;
    }
  }
  __syncthreads();
  // merge histogram into per-batch global histogram
  uint32_t* gh = ghist + (size_t)b*BINS;
  for (int i = tid; i < BINS; i += TPB){
    uint32_t c = lhist[i];
    if (c) atomicAdd(&gh[i], c);
  }
  // publish per-tile maxima (each (b,tile) written exactly once per launch)
  const int niter = (NTILES - blockIdx.x + BPB - 1) / BPB;
  if (tid < niter)
    gtmax[(size_t)b*NTILES + blockIdx.x + tid*BPB] = tmax[tid];
}

// ---------------------------------------------------------------------------
// Pass 2: find threshold bin T per batch: count(bins > T) < K <= count(bins >= T)
// ---------------------------------------------------------------------------
__global__ void thresh_kernel(const uint32_t* __restrict__ ghist, int* __restrict__ Tarr){
  __shared__ uint32_t h[BINS];
  const int b = blockIdx.x;
  for (int i = threadIdx.x; i < BINS; i += blockDim.x) h[i] = ghist[(size_t)b*BINS + i];
  __syncthreads();
  if (threadIdx.x == 0){
    uint32_t cum = 0; int t = BINS - 1;
    for (; t > 0; --t){
      uint32_t c = h[t];
      if (cum + c >= (uint32_t)TOPK) break;
      cum += c;
    }
    Tarr[b] = t;
  }
}

// ---------------------------------------------------------------------------
// Pass 3: sparse compaction. Skip tiles whose max key is below the threshold
// (~87% of reads eliminated for Gaussian data); hot tiles are re-read with
// b128 loads and candidates stored as packed (key<<32)|~idx.
// ---------------------------------------------------------------------------
__global__ void __launch_bounds__(TPB) compact_kernel(const float* __restrict__ logits,
                                                      const uint32_t* __restrict__ gtmax,
                                                      const int* __restrict__ Tarr,
                                                      uint32_t* __restrict__ cnt,
                                                      unsigned long long* __restrict__ cand){
  const int b = blockIdx.y;
  const uint32_t T = (uint32_t)Tarr[b];
  const float4* base = reinterpret_cast<const float4*>(logits) + (size_t)b*NV4;
  const uint32_t* tmx = gtmax + (size_t)b*NTILES;
  uint32_t* cb = &cnt[b];
  unsigned long long* cd = cand + (size_t)b*CAP;
  for (int t = blockIdx.x; t < NTILES; t += BPB){
    if ((tmx[t] >> 20) < T) continue;        // cold tile: 4B read instead of 4KB
    uint32_t v4 = (uint32_t)t*TPB + threadIdx.x;
    if (v4 >= (uint32_t)NV4) continue;
    float4 v = base[v4];
    uint32_t i0 = v4*4u;
    float vals[4] = {v.x, v.y, v.z, v.w};
    #pragma unroll
    for (int j = 0; j < 4; ++j){
      uint32_t key = f2key(vals[j]);
      if ((key >> 20) >= T){
        uint32_t pos = atomicAdd(cb, 1u);
        if (pos < CAP)
          cd[pos] = ((unsigned long long)key << 32) | (uint32_t)~(i0 + (uint32_t)j);
      }
    }
  }
}

// ---------------------------------------------------------------------------
// Pass 4: per-batch exact top-100 from candidates (ties -> lower index, like
// jax top_k), then sigmoid + label + box gather/scale. Writes all outputs.
// ---------------------------------------------------------------------------
__global__ void __launch_bounds__(TPB) select_kernel(const unsigned long long* __restrict__ cand,
                                                     const uint32_t* __restrict__ cnt,
                                                     const float* __restrict__ bbox,
                                                     const float* __restrict__ tsz,
                                                     float* __restrict__ out){
  __shared__ unsigned long long sc[CAP];
  __shared__ unsigned long long sr[TPB];
  __shared__ unsigned long long stopk[TOPK];
  const int b = blockIdx.x;
  const int tid = threadIdx.x;
  uint32_t n = cnt[b]; if (n > CAP) n = CAP;
  for (uint32_t i = tid; i < n; i += TPB) sc[i] = cand[(size_t)b*CAP + i];
  __syncthreads();

  for (int k = 0; k < TOPK; ++k){
    unsigned long long best = 0ull;
    for (uint32_t i = tid; i < n; i += TPB){
      unsigned long long v = sc[i];
      if (v > best) best = v;
    }
    sr[tid] = best;
    __syncthreads();
    for (int s = TPB/2; s > 0; s >>= 1){
      if (tid < s){ if (sr[tid+s] > sr[tid]) sr[tid] = sr[tid+s]; }
      __syncthreads();
    }
    unsigned long long m = sr[0];
    if (tid == 0) stopk[k] = m;
    if (m){  // packed values are unique (idx unique) -> exactly one slot cleared
      for (uint32_t i = tid; i < n; i += TPB){ if (sc[i] == m) sc[i] = 0ull; }
    }
    __syncthreads();
  }

  if (tid < TOPK){
    unsigned long long p = stopk[tid];
    uint32_t key = (uint32_t)(p >> 32);
    uint32_t idx = ~(uint32_t)p;
    float x = key2f(key);
    float score = 1.0f / (1.0f + expf(-x));
    uint32_t q = idx / (uint32_t)CLS;
    uint32_t c = idx % (uint32_t)CLS;
    if (q >= (uint32_t)QDET) q = 0;   // safety for degenerate entries
    float4 bb = reinterpret_cast<const float4*>(bbox)[(size_t)b*QDET + q];
    float ih = tsz[b*2 + 0];
    float iw = tsz[b*2 + 1];
    float hw = 0.5f*bb.z, hh = 0.5f*bb.w;
    out[(size_t)b*TOPK + tid] = score;                                   // scores
    out[(size_t)BATCH*TOPK + (size_t)b*TOPK + tid] = (float)c;           // labels
    float* bo = out + (size_t)2*BATCH*TOPK + ((size_t)b*TOPK + tid)*4;   // boxes
    bo[0] = (bb.x - hw)*iw;
    bo[1] = (bb.y - hh)*ih;
    bo[2] = (bb.x + hw)*iw;
    bo[3] = (bb.y + hh)*ih;
  }
}

// ---------------------------------------------------------------------------
extern "C" void kernel_launch(void* const* d_in, const int* in_sizes, int n_in,
                              void* d_out, int out_size, void* d_ws, size_t ws_size,
                              hipStream_t stream) {
  (void)in_sizes; (void)n_in; (void)out_size; (void)ws_size;
  const float* logits = (const float*)d_in[0];   // (64,1000,1203) f32
  const float* bbox   = (const float*)d_in[1];   // (64,1000,4)    f32
  const float* tsz    = (const float*)d_in[2];   // (64,2)         f32
  float* out = (float*)d_out;

  // Workspace layout (all offsets 8B-aligned where needed):
  //   [0)                    ghist : BATCH*BINS u32   (1 MB, zeroed per call)
  //   [+BATCH*BINS*4)        Tarr  : BATCH i32
  //   [+BATCH*4)             cnt   : BATCH u32        (zeroed per call)
  //   [+BATCH*4)             cand  : BATCH*CAP u64    (1 MB)
  //   [+BATCH*CAP*8)         gtmax : BATCH*NTILES u32 (fully rewritten per call)
  uint8_t* ws = (uint8_t*)d_ws;
  uint32_t* ghist = (uint32_t*)ws;
  int*      Tarr  = (int*)(ws + (size_t)BATCH*BINS*4);
  uint32_t* cnt   = (uint32_t*)(ws + (size_t)BATCH*BINS*4 + (size_t)BATCH*4);
  unsigned long long* cand =
      (unsigned long long*)(ws + (size_t)BATCH*BINS*4 + (size_t)2*BATCH*4);
  uint32_t* gtmax = (uint32_t*)(ws + (size_t)BATCH*BINS*4 + (size_t)2*BATCH*4
                                   + (size_t)BATCH*CAP*8);

  const int nz = BATCH*BINS + 2*BATCH;
  zero_kernel   <<<(nz + 255)/256, 256, 0, stream>>>(ghist, nz);
  hist_kernel   <<<dim3(BPB, BATCH), TPB, 0, stream>>>(logits, ghist, gtmax);
  thresh_kernel <<<BATCH, 256, 0, stream>>>(ghist, Tarr);
  compact_kernel<<<dim3(BPB, BATCH), TPB, 0, stream>>>(logits, gtmax, Tarr, cnt, cand);
  select_kernel <<<BATCH, TPB, 0, stream>>>(cand, cnt, bbox, tsz, out);
}